// DrugRank_84567906058780
// MI455X (gfx1250) — compile-verified
//
#include <hip/hip_runtime.h>
#include <math.h>

typedef __attribute__((ext_vector_type(2))) float v2f;
typedef __attribute__((ext_vector_type(8))) float v8f;

#define N_CLL 3451
#define N_MOL 64
#define N_BIO 50000
#define E_CLL 55216
#define E_MOL 128
#define E_BIO 800000

// ---------------------------------------------------------------------------
// C[M,N] = A[M,K] @ B[K,N], row-major fp32, one wave per 16x16 tile,
// V_WMMA_F32_16X16X4_F32 (f32 A/B fragments: 2 VGPRs/lane).
// A frag: row = lane%16, K = (lane/16)*2 + v   (ISA 32-bit A 16x4 layout)
// B frag: col = lane%16, K = (lane/16)*2 + v
// C/D:    VGPR v holds element (M = (lane/16)*8 + v, N = lane%16)
// Edge handling: CLAMP row/col for loads (never predicate -> no exec churn).
// Out-of-range rows/cols only pollute C elements that are masked at store.
// K must be a multiple of 4 (all layer dims here are).
// ---------------------------------------------------------------------------
__global__ __launch_bounds__(32)
void wmma_gemm_f32(const float* __restrict__ A, const float* __restrict__ B,
                   float* __restrict__ C, int M, int N, int K) {
  const int lane = threadIdx.x;
  const int half = lane >> 4;          // 0..1
  const int idx  = lane & 15;          // 0..15
  const int row  = blockIdx.y * 16 + idx;   // A row this lane loads
  const int col  = blockIdx.x * 16 + idx;   // B col this lane loads
  const int rowc = row < M ? row : M - 1;   // clamped (branch-free loads)
  const int colc = col < N ? col : N - 1;

  // per-lane base pointers (K offset half*2 folded in)
  const float* __restrict__ Ap = A + (size_t)rowc * K + half * 2;  // 8B aligned
  const float* __restrict__ Bp = B + (size_t)(half * 2) * N + colc;

  v8f acc = {};
  #pragma unroll 4
  for (int k = 0; k < K; k += 4) {
    v2f a = *(const v2f*)(Ap + k);          // one global_load_b64
    v2f b;
    b.x = Bp[(size_t)k * N];
    b.y = Bp[(size_t)(k + 1) * N];
    acc = __builtin_amdgcn_wmma_f32_16x16x4_f32(
        /*neg_a=*/false, a, /*neg_b=*/false, b,
        /*c_mod=*/(short)0, acc, /*reuse_a=*/false, /*reuse_b=*/false);
  }
  if (col < N) {
    const int rbase = blockIdx.y * 16 + half * 8;
    #pragma unroll
    for (int v = 0; v < 8; ++v) {
      const int r = rbase + v;
      if (r < M) C[(size_t)r * N + col] = acc[v];
    }
  }
}

// ---------------------------------------------------------------------------
// Graph helpers
// ---------------------------------------------------------------------------
__global__ void edge_deg(const int* __restrict__ dst, float* __restrict__ deg, int E) {
  int e = blockIdx.x * blockDim.x + threadIdx.x;
  if (e < E) atomicAdd(&deg[dst[e]], 1.0f);
}

__global__ void deg_to_dinv(float* __restrict__ dinv, const float* __restrict__ deg, int n) {
  int i = blockIdx.x * blockDim.x + threadIdx.x;
  if (i < n) dinv[i] = rsqrtf(deg[i] + 1.0f);
}

// acc[dst] += H[src] * dinv[src]*dinv[dst], flat over E*F with grid stride
__global__ void gcn_scatter(const float* __restrict__ H, const int* __restrict__ src,
                            const int* __restrict__ dst, const float* __restrict__ dinv,
                            float* __restrict__ acc, long long E, int F) {
  long long total = E * (long long)F;
  for (long long i = blockIdx.x * (long long)blockDim.x + threadIdx.x; i < total;
       i += (long long)gridDim.x * blockDim.x) {
    long long e = i / F;
    int f = (int)(i - e * F);
    int s = src[e], d = dst[e];
    atomicAdd(&acc[(size_t)d * F + f], H[(size_t)s * F + f] * dinv[s] * dinv[d]);
  }
}

// out = relu(acc + H*dinv^2 + b)   (out may alias H: pure elementwise)
__global__ void gcn_finish(float* __restrict__ out, const float* __restrict__ acc,
                           const float* __restrict__ H, const float* __restrict__ dinv,
                           const float* __restrict__ b, int n, int F) {
  long long i = blockIdx.x * (long long)blockDim.x + threadIdx.x;
  long long total = (long long)n * F;
  if (i < total) {
    int node = (int)(i / F);
    int f = (int)(i - (long long)node * F);
    float di = dinv[node];
    float v = acc[i] + H[i] * di * di + b[f];
    out[i] = fmaxf(v, 0.0f);
  }
}

// GraphConv aggregation: acc[dst] += X[src]
__global__ void sum_scatter(const float* __restrict__ X, const int* __restrict__ src,
                            const int* __restrict__ dst, float* __restrict__ acc,
                            int E, int F) {
  int i = blockIdx.x * blockDim.x + threadIdx.x;
  int total = E * F;
  if (i < total) {
    int e = i / F, f = i - e * F;
    atomicAdd(&acc[(size_t)dst[e] * F + f], X[(size_t)src[e] * F + f]);
  }
}

// out = relu(a + b + bias[f])
__global__ void add2_bias_relu(float* __restrict__ out, const float* __restrict__ a,
                               const float* __restrict__ b, const float* __restrict__ bias,
                               int n, int F) {
  int i = blockIdx.x * blockDim.x + threadIdx.x;
  if (i < n * F) {
    int f = i % F;
    out[i] = fmaxf(a[i] + b[i] + bias[f], 0.0f);
  }
}

__global__ void mean_pool(const float* __restrict__ X, float* __restrict__ y, int n, int F) {
  int f = blockIdx.x * blockDim.x + threadIdx.x;
  if (f < F) {
    float s = 0.0f;
    for (int i = 0; i < n; ++i) s += X[(size_t)i * F + f];
    y[f] = s / (float)n;
  }
}

__global__ void copy_f32(float* __restrict__ dst, const float* __restrict__ src, int n) {
  int i = blockIdx.x * blockDim.x + threadIdx.x;
  if (i < n) dst[i] = src[i];
}

// ---------------------------------------------------------------------------
// Split-K GEMV: y[j] += sum_{k in slice} x[k]*W[k*N+j]  (y pre-zeroed)
// ---------------------------------------------------------------------------
__global__ void gemv_partial(const float* __restrict__ x, const float* __restrict__ W,
                             float* __restrict__ y, int K, int N) {
  int j = blockIdx.x * blockDim.x + threadIdx.x;
  if (j >= N) return;
  int ns = gridDim.y;
  int ks = (int)(((long long)K * blockIdx.y) / ns);
  int ke = (int)(((long long)K * (blockIdx.y + 1)) / ns);
  const float* __restrict__ Wp = W + (size_t)ks * N + j;
  float s = 0.0f;
  #pragma unroll 4
  for (int k = ks; k < ke; ++k) {
    s = fmaf(x[k], *Wp, s);
    Wp += N;
  }
  atomicAdd(&y[j], s);
}

__global__ void bias_relu_vec(float* __restrict__ y, const float* __restrict__ b, int N) {
  int j = blockIdx.x * blockDim.x + threadIdx.x;
  if (j < N) y[j] = fmaxf(y[j] + b[j], 0.0f);
}

// ---------------------------------------------------------------------------
extern "C" void kernel_launch(void* const* d_in, const int* in_sizes, int n_in,
                              void* d_out, int out_size, void* d_ws, size_t ws_size,
                              hipStream_t stream) {
  // ---- inputs (setup_inputs order) ----
  const float* x_cll   = (const float*)d_in[0];
  const int*   e_cll   = (const int*)d_in[1];
  const float* x_mol   = (const float*)d_in[2];
  const int*   e_mol   = (const int*)d_in[3];
  // d_in[4] = mol_batch (all zeros) -> global mean pool
  const float* x_bio   = (const float*)d_in[5];
  const int*   e_bio   = (const int*)d_in[6];
  const float* Wc1  = (const float*)d_in[7];  const float* bc1  = (const float*)d_in[8];
  const float* Wc2  = (const float*)d_in[9];  const float* bc2  = (const float*)d_in[10];
  const float* Wc3  = (const float*)d_in[11]; const float* bc3  = (const float*)d_in[12];
  const float* Wc4  = (const float*)d_in[13]; const float* bc4  = (const float*)d_in[14];
  const float* Wl1c = (const float*)d_in[15]; const float* bl1c = (const float*)d_in[16];
  const float* Wl2c = (const float*)d_in[17]; const float* bl2c = (const float*)d_in[18];
  const float* Wl3c = (const float*)d_in[19]; const float* bl3c = (const float*)d_in[20];
  const float* Wm1r = (const float*)d_in[21]; const float* Wm1s = (const float*)d_in[22];
  const float* bm1  = (const float*)d_in[23];
  const float* Wm2r = (const float*)d_in[24]; const float* Wm2s = (const float*)d_in[25];
  const float* bm2  = (const float*)d_in[26];
  const float* Wlm  = (const float*)d_in[27]; const float* blm  = (const float*)d_in[28];
  const float* Wb1  = (const float*)d_in[29]; const float* bb1  = (const float*)d_in[30];
  const float* Wb2  = (const float*)d_in[31]; const float* bb2  = (const float*)d_in[32];
  const float* Wlb  = (const float*)d_in[33]; const float* blb  = (const float*)d_in[34];
  const float* Wd1  = (const float*)d_in[35]; const float* bd1  = (const float*)d_in[36];
  const float* Wd2  = (const float*)d_in[37]; const float* bd2  = (const float*)d_in[38];
  const float* Wcat1 = (const float*)d_in[39]; const float* bcat1 = (const float*)d_in[40];
  const float* Wcat2 = (const float*)d_in[41]; const float* bcat2 = (const float*)d_in[42];

  const int* src_cll = e_cll;           const int* dst_cll = e_cll + E_CLL;
  const int* src_mol = e_mol;           const int* dst_mol = e_mol + E_MOL;
  const int* src_bio = e_bio;           const int* dst_bio = e_bio + E_BIO;

  // ---- workspace layout (all sizes even -> 8B alignment everywhere) ----
  float* ws = (float*)d_ws;
  size_t off = 0;
  auto carve = [&](size_t n) { float* p = ws + off; off += n; return p; };
  float* buf0 = carve((size_t)N_BIO * 200);   // 10M floats
  float* buf1 = carve((size_t)N_BIO * 200);   // 10M floats
  float* deg  = carve(N_BIO);
  float* dinv = carve(N_BIO);
  float* mAgg  = carve(64 * 200);
  float* mA    = carve(64 * 200);
  float* mB    = carve(64 * 200);
  float* mC    = carve(64 * 200);
  float* mD    = carve(64 * 200);
  float* vec200  = carve(200);
  float* vecM200 = carve(200);
  float* v128m   = carve(128);
  float* v128b   = carve(128);
  float* v1000a  = carve(1000);
  float* v1000b  = carve(1000);
  float* v500    = carve(500);
  float* v256c   = carve(256);
  float* drugA   = carve(256);
  float* drugB   = carve(256);
  float* cat512  = carve(512);

  // ---- helpers ----
  auto gemm = [&](const float* A, const float* B, float* C, int M, int N, int K) {
    dim3 g((N + 15) / 16, (M + 15) / 16);
    wmma_gemm_f32<<<g, 32, 0, stream>>>(A, B, C, M, N, K);
  };
  auto gemv = [&](const float* x, const float* W, const float* b, float* y,
                  int K, int N, int splitk) {
    hipMemsetAsync(y, 0, (size_t)N * sizeof(float), stream);
    dim3 g((N + 127) / 128, splitk);
    gemv_partial<<<g, 128, 0, stream>>>(x, W, y, K, N);
    bias_relu_vec<<<(N + 127) / 128, 128, 0, stream>>>(y, b, N);
  };
  // One GCN layer: out(bufOut) = relu(propagate(X@W) + b).
  // H -> bufH, acc -> bufAcc (X's buffer may be reused as bufAcc after gemm;
  // bufOut may alias bufH since gcn_finish is elementwise).
  auto gcn = [&](const float* X, const float* W, const float* b,
                 const int* src, const int* dst, const float* dv,
                 int n, int E, int Kin, int F,
                 float* bufH, float* bufAcc, float* bufOut) {
    gemm(X, W, bufH, n, F, Kin);
    hipMemsetAsync(bufAcc, 0, (size_t)n * F * sizeof(float), stream);
    long long total = (long long)E * F;
    long long want = (total + 255) / 256;
    int blocks = (int)(want > 1048576 ? 1048576 : want);
    gcn_scatter<<<blocks, 256, 0, stream>>>(bufH, src, dst, dv, bufAcc, (long long)E, F);
    long long nf = (long long)n * F;
    gcn_finish<<<(int)((nf + 255) / 256), 256, 0, stream>>>(bufOut, bufAcc, bufH, dv, b, n, F);
  };

  // ========================= bio branch =========================
  hipMemsetAsync(deg, 0, N_BIO * sizeof(float), stream);
  edge_deg<<<(E_BIO + 255) / 256, 256, 0, stream>>>(dst_bio, deg, E_BIO);
  deg_to_dinv<<<(N_BIO + 255) / 256, 256, 0, stream>>>(dinv, deg, N_BIO);
  // layer 1: x_bio[50000,256] -> buf0
  gcn(x_bio, Wb1, bb1, src_bio, dst_bio, dinv, N_BIO, E_BIO, 256, 200, buf0, buf1, buf0);
  // layer 2: buf0 -> buf1 (acc reuses buf0 after gemm)
  gcn(buf0, Wb2, bb2, src_bio, dst_bio, dinv, N_BIO, E_BIO, 200, 200, buf1, buf0, buf1);
  // last node row -> vec200 ; dense 200->128
  copy_f32<<<2, 128, 0, stream>>>(vec200, buf1 + (size_t)(N_BIO - 1) * 200, 200);
  gemv(vec200, Wlb, blb, v128b, 200, 128, 2);

  // ========================= molecule branch =========================
  // GraphConv 1: agg = scatter(x_mol); m = relu(agg@Wm1r + x@Wm1s + bm1)
  hipMemsetAsync(mAgg, 0, 64 * 64 * sizeof(float), stream);
  sum_scatter<<<(E_MOL * 64 + 255) / 256, 256, 0, stream>>>(x_mol, src_mol, dst_mol, mAgg, E_MOL, 64);
  gemm(mAgg, Wm1r, mA, 64, 200, 64);
  gemm(x_mol, Wm1s, mB, 64, 200, 64);
  add2_bias_relu<<<(64 * 200 + 255) / 256, 256, 0, stream>>>(mC, mA, mB, bm1, 64, 200);
  // GraphConv 2
  hipMemsetAsync(mAgg, 0, 64 * 200 * sizeof(float), stream);
  sum_scatter<<<(E_MOL * 200 + 255) / 256, 256, 0, stream>>>(mC, src_mol, dst_mol, mAgg, E_MOL, 200);
  gemm(mAgg, Wm2r, mA, 64, 200, 200);
  gemm(mC, Wm2s, mB, 64, 200, 200);
  add2_bias_relu<<<(64 * 200 + 255) / 256, 256, 0, stream>>>(mD, mA, mB, bm2, 64, 200);
  // global mean pool + dense 200->128
  mean_pool<<<2, 128, 0, stream>>>(mD, vecM200, 64, 200);
  gemv(vecM200, Wlm, blm, v128m, 200, 128, 2);

  // ========================= cell-line branch =========================
  hipMemsetAsync(deg, 0, N_CLL * sizeof(float), stream);
  edge_deg<<<(E_CLL + 255) / 256, 256, 0, stream>>>(dst_cll, deg, E_CLL);
  deg_to_dinv<<<(N_CLL + 255) / 256, 256, 0, stream>>>(dinv, deg, N_CLL);
  gcn(x_cll, Wc1, bc1, src_cll, dst_cll, dinv, N_CLL, E_CLL, 512, 200, buf0, buf1, buf0);
  gcn(buf0, Wc2, bc2, src_cll, dst_cll, dinv, N_CLL, E_CLL, 200, 200, buf1, buf0, buf1);
  gcn(buf1, Wc3, bc3, src_cll, dst_cll, dinv, N_CLL, E_CLL, 200, 200, buf0, buf1, buf0);
  gcn(buf0, Wc4, bc4, src_cll, dst_cll, dinv, N_CLL, E_CLL, 200, 3,   buf1, buf0, buf1);
  // buf1 now holds [3451,3] row-major == flattened [1, 10353]
  gemv(buf1,   Wl1c, bl1c, v1000a, 3 * N_CLL, 1000, 32);
  gemv(v1000a, Wl2c, bl2c, v1000b, 1000, 1000, 8);
  gemv(v1000b, Wl3c, bl3c, v256c,  1000, 256, 8);

  // ========================= fusion head =========================
  copy_f32<<<1, 128, 0, stream>>>(drugA,       v128m, 128);
  copy_f32<<<1, 128, 0, stream>>>(drugA + 128, v128b, 128);
  gemv(drugA, Wd1, bd1, v500,  256, 500, 2);
  gemv(v500,  Wd2, bd2, drugB, 500, 256, 4);
  copy_f32<<<2, 128, 0, stream>>>(cat512,       drugB, 256);
  copy_f32<<<2, 128, 0, stream>>>(cat512 + 256, v256c, 256);
  gemv(cat512, Wcat1, bcat1, v1000a, 512, 1000, 4);
  gemv(v1000a, Wcat2, bcat2, (float*)d_out, 1000, 1, 8);
}